// PtrNet2_87617332838641
// MI455X (gfx1250) — compile-verified
//
#include <hip/hip_runtime.h>
#include <hip/hip_bf16.h>
#include <math.h>

typedef float v2f __attribute__((ext_vector_type(2)));
typedef float v8f __attribute__((ext_vector_type(8)));

#define B_SZ   256
#define N_FULL 1026
#define N_REF  1024
#define H      128

// LDS layout (floats):
//  A tile: [128][LDA] row-major [m][k]; LDA%64==4 -> A-frag b64 reads use banks 4l+{0..3}: conflict-free
//  B tile: k-pair interleaved: elem (k,h) at (k>>1)*BPS + 2h + (k&1);
//          BPS%64==32 -> lane halves (k vs k+2) are 32 banks apart: all 64 banks hit once per b64 frag load
#define LDA 132
#define BPS 288
#define LDS_FLOATS (128 * LDA + 64 * BPS)

__device__ __forceinline__ float tanh_fast(float x) {
#if __has_builtin(__builtin_amdgcn_tanhf)
    return __builtin_amdgcn_tanhf(x);
#elif __has_builtin(__builtin_amdgcn_tanh_f32)
    return __builtin_amdgcn_tanh_f32(x);
#else
    float cx = fminf(fmaxf(x, -9.0f), 9.0f);
    float e  = __expf(2.0f * cx);
    return (e - 1.0f) * __builtin_amdgcn_rcpf(e + 1.0f);
#endif
}

// ---------------------------------------------------------------------------
// query0[b,h] = enc_h[b, N-2, h]
// ---------------------------------------------------------------------------
__global__ void ptr_initq(const float* __restrict__ enc, float* __restrict__ query) {
    int b = blockIdx.x, h = threadIdx.x;
    query[b * H + h] = enc[((size_t)b * N_FULL + (N_FULL - 2)) * H + h];
}

// ---------------------------------------------------------------------------
// u2[b,n,h] = sum_k ref[b,n,k] * Wref_w[h,k] + Wref_b[h]
// 256-thread block -> 128(M) x 128(H) tile; 8 waves x (16-row M tile, 8 H tiles)
// via V_WMMA_F32_16X16X4_F32, K fully unrolled (32 steps of 4) so the scheduler
// can hoist ds_load_b64 fragment loads across WMMA bursts.
// ---------------------------------------------------------------------------
__global__ __launch_bounds__(256, 1) void ptr_u2_gemm(
    const float* __restrict__ enc, const float* __restrict__ Wref_w,
    const float* __restrict__ Wref_b, float* __restrict__ u2) {
    extern __shared__ float smem[];
    float* ldsA = smem;               // [128][LDA]
    float* ldsB = smem + 128 * LDA;   // [64 k-pairs][BPS]

    const int tid   = threadIdx.x;
    const int block = blockIdx.x;            // 2048 blocks
    const int b     = block >> 3;            // 8 blocks per batch (1024/128)
    const int n0    = (block & 7) * 128;
    const float* Abase = enc + ((size_t)b * N_FULL + n0) * H;

    // Stage A tile (coalesced float4 loads)
    for (int idx = tid; idx < 128 * 32; idx += 256) {
        int r = idx >> 5, c4 = (idx & 31) * 4;
        const float4 v = *(const float4*)(Abase + (size_t)r * H + c4);
        float* dst = ldsA + r * LDA + c4;
        dst[0] = v.x; dst[1] = v.y; dst[2] = v.z; dst[3] = v.w;
    }
    // Stage B = Wref_w^T, k-pair interleaved (coalesced global read)
    for (int idx = tid; idx < 128 * 128; idx += 256) {
        int h = idx >> 7, k = idx & 127;
        ldsB[(k >> 1) * BPS + 2 * h + (k & 1)] = Wref_w[idx];
    }
    __syncthreads();

    const int wave = tid >> 5;
    const int lane = tid & 31;
    const int half = lane >> 4;   // K sub-offset select (ISA 16x4 f32 A/B layout)
    const int lrow = lane & 15;   // M row (A) / N column (B, C, D)
    const int m0   = wave * 16;

    v8f acc[8];
#pragma unroll
    for (int j = 0; j < 8; ++j) {
        float bias = Wref_b[j * 16 + lrow];
        acc[j] = {bias, bias, bias, bias, bias, bias, bias, bias};
    }

    const float* aptr = ldsA + (m0 + lrow) * LDA + half * 2;
    const float* bptr = ldsB + half * BPS + 2 * lrow;
#pragma unroll
    for (int k0 = 0; k0 < 128; k0 += 4) {
        const v2f a = *(const v2f*)(aptr + k0);
        const float* bbase = bptr + (k0 >> 1) * BPS;   // (k0>>1)+half k-pair row
        v2f bf[8];
#pragma unroll
        for (int j = 0; j < 8; ++j) bf[j] = *(const v2f*)(bbase + j * 32);
#pragma unroll
        for (int j = 0; j < 8; ++j)
            acc[j] = __builtin_amdgcn_wmma_f32_16x16x4_f32(
                false, a, false, bf[j], (short)0, acc[j], false, false);
    }

    // Store: C/D layout -> VGPR r holds (M=r, N=lane) | (M=r+8, N=lane-16)
    float* out = u2 + ((size_t)b * N_REF + n0 + m0) * H;
#pragma unroll
    for (int j = 0; j < 8; ++j) {
#pragma unroll
        for (int r = 0; r < 8; ++r) {
            int m = r + half * 8;
            out[(size_t)m * H + j * 16 + lrow] = acc[j][r];
        }
    }
}

// ---------------------------------------------------------------------------
// One glimpse iteration. One block per batch element b, 256 threads (8 waves).
// ---------------------------------------------------------------------------
__global__ __launch_bounds__(256, 1) void ptr_glimpse(
    const float* __restrict__ enc, const float* __restrict__ u2,
    const float* __restrict__ Vec, const float* __restrict__ Wq_w,
    const float* __restrict__ Wq_b, float* __restrict__ query) {
    __shared__ float s_q[H];
    __shared__ float s_u1[H];
    __shared__ float s_u[N_REF];
    __shared__ float s_red[256];
    __shared__ float s_acc[8 * H];

    const int b = blockIdx.x, tid = threadIdx.x;

    if (tid < H) s_q[tid] = query[b * H + tid];
    __syncthreads();

    // u1 = query @ Wq_w^T + Wq_b
    if (tid < H) {
        float s = Wq_b[tid];
        const float* wr = Wq_w + tid * H;
        for (int k = 0; k < H; ++k) s = fmaf(s_q[k], wr[k], s);
        s_u1[tid] = s;
    }
    __syncthreads();

    // u[n] = sum_h tanh(u1[h] + u2[b,n,h]) * Vec[h]
    // wave per n; each lane covers 4 contiguous h -> one b128 load per row/lane
    const int wave = tid >> 5, lane = tid & 31;
    const float4 u1v = *(const float4*)&s_u1[lane * 4];
    const float4 vv  = *(const float4*)&Vec[lane * 4];
    const float* u2b = u2 + (size_t)b * N_REF * H;
#pragma unroll 2
    for (int n = wave; n < N_REF; n += 8) {
        const float4 r4 = *(const float4*)(u2b + (size_t)n * H + lane * 4);
        float p;
        p = tanh_fast(u1v.x + r4.x) * vv.x;
        p = fmaf(tanh_fast(u1v.y + r4.y), vv.y, p);
        p = fmaf(tanh_fast(u1v.z + r4.z), vv.z, p);
        p = fmaf(tanh_fast(u1v.w + r4.w), vv.w, p);
#pragma unroll
        for (int off = 16; off > 0; off >>= 1) p += __shfl_xor(p, off, 32);
        if (lane == 0) s_u[n] = p;
    }
    __syncthreads();

    // softmax over n
    float m = -1e30f;
    for (int n = tid; n < N_REF; n += 256) m = fmaxf(m, s_u[n]);
    s_red[tid] = m;
    __syncthreads();
    for (int s = 128; s > 0; s >>= 1) {
        if (tid < s) s_red[tid] = fmaxf(s_red[tid], s_red[tid + s]);
        __syncthreads();
    }
    const float mx = s_red[0];
    __syncthreads();
    float sum = 0.0f;
    for (int n = tid; n < N_REF; n += 256) {
        float e = __expf(s_u[n] - mx);
        s_u[n] = e;
        sum += e;
    }
    s_red[tid] = sum;
    __syncthreads();
    for (int s = 128; s > 0; s >>= 1) {
        if (tid < s) s_red[tid] += s_red[tid + s];
        __syncthreads();
    }
    const float inv = 1.0f / s_red[0];
    __syncthreads();

    // new query[b,h] = inv * sum_n e[n] * ref[b,n,h]   (float4 over h, waves split n)
    const int seg = tid >> 5;           // 8 segments of 128 n each
    const int h4  = (tid & 31) * 4;
    const float* refb = enc + (size_t)b * N_FULL * H;
    float ax = 0.f, ay = 0.f, az = 0.f, aw = 0.f;
    for (int n = seg * 128; n < seg * 128 + 128; ++n) {
        const float  w  = s_u[n];
        const float4 r4 = *(const float4*)(refb + (size_t)n * H + h4);
        ax = fmaf(w, r4.x, ax); ay = fmaf(w, r4.y, ay);
        az = fmaf(w, r4.z, az); aw = fmaf(w, r4.w, aw);
    }
    s_acc[seg * H + h4 + 0] = ax;
    s_acc[seg * H + h4 + 1] = ay;
    s_acc[seg * H + h4 + 2] = az;
    s_acc[seg * H + h4 + 3] = aw;
    __syncthreads();
    if (tid < H) {
        float s = 0.0f;
#pragma unroll
        for (int sg = 0; sg < 8; ++sg) s += s_acc[sg * H + tid];
        query[b * H + tid] = s * inv;
    }
}

// ---------------------------------------------------------------------------
// pred[b] = relu(query[b] @ W1^T) @ W2^T
// ---------------------------------------------------------------------------
__global__ void ptr_head(const float* __restrict__ query, const float* __restrict__ W1,
                         const float* __restrict__ W2, float* __restrict__ out) {
    __shared__ float sq[H];
    __shared__ float red[H];
    const int b = blockIdx.x, h = threadIdx.x;
    sq[h] = query[b * H + h];
    __syncthreads();
    float s = 0.0f;
    const float* w = W1 + h * H;
    for (int k = 0; k < H; ++k) s = fmaf(sq[k], w[k], s);
    red[h] = fmaxf(s, 0.0f) * W2[h];
    __syncthreads();
    for (int st = 64; st > 0; st >>= 1) {
        if (h < st) red[h] += red[h + st];
        __syncthreads();
    }
    if (h == 0) out[b] = red[0];
}

extern "C" void kernel_launch(void* const* d_in, const int* in_sizes, int n_in,
                              void* d_out, int out_size, void* d_ws, size_t ws_size,
                              hipStream_t stream) {
    (void)in_sizes; (void)n_in; (void)out_size; (void)ws_size;
    const float* enc    = (const float*)d_in[0];
    const float* Vec    = (const float*)d_in[1];
    const float* Wq_w   = (const float*)d_in[2];
    const float* Wq_b   = (const float*)d_in[3];
    const float* Wref_w = (const float*)d_in[4];
    const float* Wref_b = (const float*)d_in[5];
    const float* W1     = (const float*)d_in[6];
    const float* W2     = (const float*)d_in[7];
    float* out = (float*)d_out;

    float* u2    = (float*)d_ws;                                        // 256*1024*128 f32
    float* query = (float*)((char*)d_ws +
                            (size_t)B_SZ * N_REF * H * sizeof(float));  // 256*128 f32

    const int ldsBytes = LDS_FLOATS * (int)sizeof(float);
    hipFuncSetAttribute((const void*)ptr_u2_gemm,
                        hipFuncAttributeMaxDynamicSharedMemorySize, ldsBytes);

    ptr_initq<<<B_SZ, H, 0, stream>>>(enc, query);
    ptr_u2_gemm<<<(B_SZ * N_REF) / 128, 256, ldsBytes, stream>>>(enc, Wref_w, Wref_b, u2);
    for (int it = 0; it < 3; ++it)
        ptr_glimpse<<<B_SZ, 256, 0, stream>>>(enc, u2, Vec, Wq_w, Wq_b, query);
    ptr_head<<<B_SZ, H, 0, stream>>>(query, W1, W2, out);
}